// L_exp_classes_4166118277840
// MI455X (gfx1250) — compile-verified
//
#include <hip/hip_runtime.h>
#include <cstdint>

#define OUT      512
#define TILE     32
#define HIN      128
#define CLAB     19
#define PATCH    10
#define PATCH_EL (CLAB * PATCH * PATCH)   // 1900 floats = 7.6 KB
#define VROW_W   (CLAB * PATCH)           // 190 interpolated values per row
#define VROW_PAD 192                      // per-wave scratch stride
#define SCALE    (127.0f / 511.0f)        // align_corners: (H-1)/(OUT-1)

// ---- CDNA5 async global->LDS staging -------------------------------------
__device__ __forceinline__ void async_ld_f32(const float* g, float* l) {
#if __has_builtin(__builtin_amdgcn_global_load_async_to_lds_b32)
  __builtin_amdgcn_global_load_async_to_lds_b32(
      (__attribute__((address_space(1))) int*)(uintptr_t)g,
      (__attribute__((address_space(3))) int*)(uintptr_t)(uint32_t)(uintptr_t)l,
      0, 0);
#else
  uint32_t loff = (uint32_t)(uintptr_t)l;
  asm volatile("global_load_async_to_lds_b32 %0, %1, off"
               :: "v"(loff), "v"((uint64_t)(uintptr_t)g)
               : "memory");
#endif
}

__device__ __forceinline__ void wait_async_zero() {
  asm volatile("s_wait_asynccnt 0x0" ::: "memory");
}

// ---- pass 1: fused upsample + argmax + gather + squared-error partials ---
__global__ __launch_bounds__(256) void fused_mse_kernel(
    const float* __restrict__ x,      // (16,3,512,512)
    const float* __restrict__ label,  // (16,19,128,128)
    const float* __restrict__ cls,    // (19)
    float* __restrict__ partial)      // (4096) block partial sums
{
  __shared__ float s_lab[PATCH_EL];        // 19 x 10 x 10 border-duplicated patch
  __shared__ float s_vrow[8 * VROW_PAD];   // per-wave row-interp scratch
  __shared__ float s_cls[CLAB];
  __shared__ float s_red[8];

  const int tid = threadIdx.x;
  const int tx  = blockIdx.x, ty = blockIdx.y, b = blockIdx.z;
  const int ox0 = tx * TILE,  oy0 = ty * TILE;

  const int x_base = (int)(ox0 * SCALE);   // non-negative -> trunc == floor
  const int y_base = (int)(oy0 * SCALE);

  // Stage the 19x10x10 label patch into LDS asynchronously. min() clamping
  // duplicates the border row/col, so the compute path never needs to clamp.
  const float* lab_b = label + (size_t)b * CLAB * HIN * HIN;
  for (int i = tid; i < PATCH_EL; i += 256) {
    int c   = i / (PATCH * PATCH);
    int rem = i - c * PATCH * PATCH;
    int r   = rem / PATCH;
    int col = rem - r * PATCH;
    int gy  = min(y_base + r,   HIN - 1);
    int gx  = min(x_base + col, HIN - 1);
    async_ld_f32(lab_b + ((size_t)c * HIN + gy) * HIN + gx, &s_lab[i]);
  }
  if (tid < CLAB) s_cls[tid] = cls[tid];
  wait_async_zero();      // drain this wave's ASYNCcnt
  __syncthreads();        // make all waves' LDS writes visible

  const int wave = tid >> 5, lane = tid & 31;
  float* vrow = &s_vrow[wave * VROW_PAD];

  // Per-lane horizontal constants (fixed across all 4 rows this lane handles)
  const int   ox  = ox0 + lane;
  const float xsf = ox * SCALE;
  const int   xq  = (int)xsf;
  const float wx  = xsf - (float)xq;
  const float wl  = 1.f - wx;
  const int   rx0 = xq - x_base;           // rx1 = rx0 + 1 (border duplicated)

  const size_t plane = (size_t)OUT * OUT;
  const float* xb = x + (size_t)b * 3 * plane;

  float acc = 0.f;
#pragma unroll
  for (int k = 0; k < 4; ++k) {
    // All lanes of this wave share output row ly; build its v-interp once.
    const int   ly  = wave + 8 * k;
    const int   oy  = oy0 + ly;
    const float ysf = oy * SCALE;
    const int   y0  = (int)ysf;
    const float wy  = ysf - (float)y0;
    const float wt  = 1.f - wy;
    const int   ry0 = y0 - y_base;         // ry1 = ry0 + 1 (border duplicated)

    // vrow[c*10+col] = (1-wy)*lab[c][ry0][col] + wy*lab[c][ry0+1][col]
    // Per-wave private region: no barrier; LDS is in-order within a wave.
    for (int j = lane; j < VROW_W; j += 32) {
      const int c   = j / PATCH;
      const int col = j - c * PATCH;
      const int idx = c * (PATCH * PATCH) + ry0 * PATCH + col;
      vrow[j] = wt * s_lab[idx] + wy * s_lab[idx + PATCH];   // offsets 0,+10: 2addr
    }

    // Horizontal interp + argmax over 19 channels (2 LDS reads / channel,
    // immediate offsets c*10 and c*10+1 -> ds_load_2addr_b32 candidates).
    const float* vb = vrow + rx0;
    float best = -__builtin_inff();
    int   bidx = 0;
#pragma unroll
    for (int c = 0; c < CLAB; ++c) {
      const float a = vb[c * PATCH];
      const float bb = vb[c * PATCH + 1];
      const float v = wl * a + wx * bb;    // same assoc. as reference
      if (v > best) { best = v; bidx = c; }   // strict '>' == first-max
    }

    // Channel-mean of x: streamed once -> non-temporal (preserve L2 for label)
    const size_t off = (size_t)oy * OUT + ox;
    const float  x0v = __builtin_nontemporal_load(xb + off);
    const float  x1v = __builtin_nontemporal_load(xb + plane + off);
    const float  x2v = __builtin_nontemporal_load(xb + 2 * plane + off);
    const float  xm  = (x0v + x1v + x2v) * (1.0f / 3.0f);
    const float  d   = xm - s_cls[bidx];
    acc += d * d;
  }

  // wave32 tree reduce, then cross-wave via LDS (deterministic)
  for (int o = 16; o > 0; o >>= 1) acc += __shfl_down(acc, o, 32);
  if (lane == 0) s_red[wave] = acc;
  __syncthreads();
  if (wave == 0) {
    float v = (lane < 8) ? s_red[lane] : 0.f;
    for (int o = 4; o > 0; o >>= 1) v += __shfl_down(v, o, 32);
    if (lane == 0) partial[(b * 16 + ty) * 16 + tx] = v;
  }
}

// ---- pass 2: deterministic final reduction of 4096 partials --------------
__global__ __launch_bounds__(256) void final_reduce_kernel(
    const float* __restrict__ partial, float* __restrict__ out)
{
  __shared__ float s_red[8];
  const int tid = threadIdx.x;
  float acc = 0.f;
  for (int i = tid; i < 4096; i += 256) acc += partial[i];
  for (int o = 16; o > 0; o >>= 1) acc += __shfl_down(acc, o, 32);
  const int wave = tid >> 5, lane = tid & 31;
  if (lane == 0) s_red[wave] = acc;
  __syncthreads();
  if (wave == 0) {
    float v = (lane < 8) ? s_red[lane] : 0.f;
    for (int o = 4; o > 0; o >>= 1) v += __shfl_down(v, o, 32);
    if (lane == 0) out[0] = v * (1.0f / (16.0f * 512.0f * 512.0f));
  }
}

extern "C" void kernel_launch(void* const* d_in, const int* in_sizes, int n_in,
                              void* d_out, int out_size, void* d_ws, size_t ws_size,
                              hipStream_t stream) {
  const float* x     = (const float*)d_in[0];   // (16,3,512,512) f32
  const float* label = (const float*)d_in[1];   // (16,19,128,128) f32
  const float* cls   = (const float*)d_in[2];   // (19,) f32
  float* partial = (float*)d_ws;                // 4096 floats of scratch
  float* out     = (float*)d_out;               // 1 float

  dim3 grid(OUT / TILE, OUT / TILE, 16);        // 16x16 tiles x 16 batches
  dim3 block(256);                              // 8 wave32 waves
  fused_mse_kernel<<<grid, block, 0, stream>>>(x, label, cls, partial);
  final_reduce_kernel<<<dim3(1), block, 0, stream>>>(partial, out);
}